// GFM_69818988364106
// MI455X (gfx1250) — compile-verified
//
#include <hip/hip_runtime.h>
#include <math.h>

// ---------------------------------------------------------------------------
// Problem constants (from reference): x (16,256,128,128) f32, conv_w (64,256)
// ---------------------------------------------------------------------------
#define B_    16
#define CIN   256
#define COUT  64
#define HW    16384          // 128*128
#define NPM   (64 * HW)      // elements per modality = 1,048,576
#define NB    128            // output columns per GEMM block
#define WROW  257            // LDS row stride for weights (bank-conflict pad)

typedef float v2f __attribute__((ext_vector_type(2)));
typedef float v8f __attribute__((ext_vector_type(8)));

__device__ __forceinline__ float dot4(float4 a, float4 b) {
    return a.x * b.x + a.y * b.y + a.z * b.z + a.w * b.w;
}

// ---------------------------------------------------------------------------
// Kernel 0: zero the gram scratch (ws is poisoned with 0xAA, not zero)
// ---------------------------------------------------------------------------
__global__ void zero_kernel(float* __restrict__ g, int n) {
    int i = blockIdx.x * blockDim.x + threadIdx.x;
    if (i < n) g[i] = 0.0f;
}

// ---------------------------------------------------------------------------
// Kernel 1: pairwise modality dot products. 10 unique (i<=j) accumulators.
// Grid: 16 b * 64 chunks. Each block streams 16384 floats per modality
// (float4 vectorized), wave-reduces, atomically adds into gram[b*10 + k].
// ---------------------------------------------------------------------------
__global__ __launch_bounds__(256) void gram_kernel(const float* __restrict__ x,
                                                   float* __restrict__ gram) {
    const int b = blockIdx.x >> 6;
    const int chunk = blockIdx.x & 63;
    const size_t Nq = NPM / 4;                                // float4 per modality
    const float4* xb = reinterpret_cast<const float4*>(x) + (size_t)b * 4 * Nq;
    const float4* m0 = xb;
    const float4* m1 = xb + Nq;
    const float4* m2 = xb + 2 * Nq;
    const float4* m3 = xb + 3 * Nq;

    size_t beg = (size_t)chunk * (Nq / 64) + threadIdx.x;     // 4096 float4 / chunk
    float a[10];
#pragma unroll
    for (int j = 0; j < 10; ++j) a[j] = 0.0f;

#pragma unroll 4
    for (int it = 0; it < 16; ++it) {
        size_t i = beg + (size_t)it * 256;
        float4 v0 = m0[i], v1 = m1[i], v2 = m2[i], v3 = m3[i];
        a[0] += dot4(v0, v0);
        a[1] += dot4(v0, v1);
        a[2] += dot4(v0, v2);
        a[3] += dot4(v0, v3);
        a[4] += dot4(v1, v1);
        a[5] += dot4(v1, v2);
        a[6] += dot4(v1, v3);
        a[7] += dot4(v2, v2);
        a[8] += dot4(v2, v3);
        a[9] += dot4(v3, v3);
    }
    // wave32 shuffle reduction
#pragma unroll
    for (int off = 16; off; off >>= 1)
#pragma unroll
        for (int j = 0; j < 10; ++j) a[j] += __shfl_xor(a[j], off, 32);

    if ((threadIdx.x & 31) == 0)
#pragma unroll
        for (int j = 0; j < 10; ++j) atomicAdd(&gram[b * 10 + j], a[j]);
}

// ---------------------------------------------------------------------------
// Kernel 2: cosine matrix -> attention softmax. One thread per batch.
// score_i = -( sum_j G(i,j)/(nrm_i*nrm_j) - 1 )   (exact reference formula,
// including the cos(i,i) term with the 1e-8 norm clamp)
// ---------------------------------------------------------------------------
__global__ void attn_kernel(const float* __restrict__ gram, float* __restrict__ prob) {
    int b = threadIdx.x;
    if (b >= B_) return;
    const float* g = gram + b * 10;
    auto fetch = [&](int i, int j) -> float {
        if (i > j) { int t = i; i = j; j = t; }
        return g[i * 4 - (i * (i + 1)) / 2 + j];              // upper-tri index
    };
    float nrm[4];
#pragma unroll
    for (int i = 0; i < 4; ++i) nrm[i] = fmaxf(sqrtf(fetch(i, i)), 1e-8f);

    float s[4];
#pragma unroll
    for (int i = 0; i < 4; ++i) {
        float acc = 0.0f;
#pragma unroll
        for (int j = 0; j < 4; ++j) acc += fetch(i, j) / (nrm[i] * nrm[j]);
        s[i] = -(acc - 1.0f);
    }
    float mx = fmaxf(fmaxf(s[0], s[1]), fmaxf(s[2], s[3]));
    float e[4], sum = 0.0f;
#pragma unroll
    for (int i = 0; i < 4; ++i) { e[i] = expf(s[i] - mx); sum += e[i]; }
#pragma unroll
    for (int i = 0; i < 4; ++i) prob[b * 4 + i] = e[i] / sum;
}

// ---------------------------------------------------------------------------
// Kernel 3: 1x1 conv as WMMA fp32 GEMM.  Per block: one b, a 64x128 Y tile.
//   Y[64 x NB] = (W * prob)[64 x 256] @ X[256 x NB]
// Weights (attention-scaled) staged in LDS with stride-257 padding.
// 8 waves: wave w owns columns [w*16, w*16+16) and all 4 M-tiles.
// WMMA f32 16x16x4 fragment layouts (ISA 7.12.2):
//   A (16x4):  lanes 0-15 M=lane  {v0=K0, v1=K1}; lanes 16-31 {v0=K2, v1=K3}
//   B (4x16):  lanes 0-15 N=lane  {v0=K0, v1=K1}; lanes 16-31 {v0=K2, v1=K3}
//   C (16x16): lanes 0-15 N=lane, VGPR r = M=r ; lanes 16-31 M=8+r
// EXEC is all-1s: 256-thread blocks, no divergence in this kernel.
// ---------------------------------------------------------------------------
__global__ __launch_bounds__(256) void gemm_kernel(const float* __restrict__ x,
                                                   const float* __restrict__ w,
                                                   const float* __restrict__ prob,
                                                   float* __restrict__ y) {
    __shared__ float sW[COUT * WROW];                        // ~64.25 KB

    const int b  = blockIdx.x >> 7;                          // HW/NB = 128 blocks per b
    const int n0 = (blockIdx.x & 127) * NB;
    const int tid = threadIdx.x;

    float p0 = prob[b * 4 + 0], p1 = prob[b * 4 + 1];
    float p2 = prob[b * 4 + 2], p3 = prob[b * 4 + 3];

    // Stage attention-scaled weights into LDS (64*256 = 16384 floats, 64/thread)
    for (int i = tid; i < COUT * CIN; i += 256) {
        int o = i >> 8, c = i & 255;
        int g = c >> 6;
        float pv = (g == 0) ? p0 : (g == 1) ? p1 : (g == 2) ? p2 : p3;
        sW[o * WROW + c] = w[i] * pv;
    }
    __syncthreads();

    const int wave = tid >> 5;
    const int lane = tid & 31;
    const int half = lane >> 4;                              // 0: K0/K1, 1: K2/K3
    const int l16  = lane & 15;
    const int kofs = half * 2;
    const int n    = n0 + wave * 16 + l16;                   // this lane's column

    const float* xc = x + (size_t)b * CIN * HW + n;          // + k*HW per K row
    v8f acc[4] = {};                                         // 4 M-tiles

#pragma unroll 4
    for (int k = 0; k < CIN; k += 4) {
        v2f bf;                                              // B fragment (X)
        bf.x = xc[(size_t)(k + kofs) * HW];
        bf.y = xc[(size_t)(k + kofs + 1) * HW];
#pragma unroll
        for (int mt = 0; mt < 4; ++mt) {
            int m = mt * 16 + l16;
            v2f af;                                          // A fragment (W)
            af.x = sW[m * WROW + k + kofs];
            af.y = sW[m * WROW + k + kofs + 1];
            acc[mt] = __builtin_amdgcn_wmma_f32_16x16x4_f32(
                false, af, false, bf, (short)0, acc[mt], false, false);
        }
    }

    // Store C tiles: lane's column n, 8 rows per half-wave -> coalesced 64B runs
    float* yb = y + (size_t)b * COUT * HW + n;
#pragma unroll
    for (int mt = 0; mt < 4; ++mt)
#pragma unroll
        for (int r = 0; r < 8; ++r) {
            int o = mt * 16 + half * 8 + r;
            yb[(size_t)o * HW] = acc[mt][r];
        }
}

// ---------------------------------------------------------------------------
// Kernel 4: InstanceNorm (affine=False, eps=1e-5) + ReLU, in place on d_out.
// One block per (b,o): reduce 16384 values, then normalize (all L2-hot).
// ---------------------------------------------------------------------------
__global__ __launch_bounds__(256) void inorm_kernel(float* __restrict__ y) {
    float4* yv = reinterpret_cast<float4*>(y + (size_t)blockIdx.x * HW);
    const int tid = threadIdx.x;

    float s = 0.0f, s2 = 0.0f;
#pragma unroll 4
    for (int i = tid; i < HW / 4; i += 256) {
        float4 v = yv[i];
        s  += v.x + v.y + v.z + v.w;
        s2 += v.x * v.x + v.y * v.y + v.z * v.z + v.w * v.w;
    }
#pragma unroll
    for (int off = 16; off; off >>= 1) {
        s  += __shfl_xor(s, off, 32);
        s2 += __shfl_xor(s2, off, 32);
    }
    __shared__ float rs[8], rs2[8];
    __shared__ float smean, sinv;
    const int wave = tid >> 5, lane = tid & 31;
    if (lane == 0) { rs[wave] = s; rs2[wave] = s2; }
    __syncthreads();
    if (tid == 0) {
        float S = 0.0f, S2 = 0.0f;
        for (int i = 0; i < 8; ++i) { S += rs[i]; S2 += rs2[i]; }
        float mean = S * (1.0f / HW);
        float var  = S2 * (1.0f / HW) - mean * mean;
        smean = mean;
        sinv  = rsqrtf(var + 1e-5f);
    }
    __syncthreads();
    const float mean = smean, inv = sinv;
#pragma unroll 4
    for (int i = tid; i < HW / 4; i += 256) {
        float4 v = yv[i];
        v.x = fmaxf((v.x - mean) * inv, 0.0f);
        v.y = fmaxf((v.y - mean) * inv, 0.0f);
        v.z = fmaxf((v.z - mean) * inv, 0.0f);
        v.w = fmaxf((v.w - mean) * inv, 0.0f);
        yv[i] = v;
    }
}

// ---------------------------------------------------------------------------
// Host launcher
// ---------------------------------------------------------------------------
extern "C" void kernel_launch(void* const* d_in, const int* in_sizes, int n_in,
                              void* d_out, int out_size, void* d_ws, size_t ws_size,
                              hipStream_t stream) {
    const float* x = (const float*)d_in[0];   // (16,256,128,128)
    const float* w = (const float*)d_in[1];   // (64,256)
    float* out  = (float*)d_out;              // (16,64,128,128)
    float* gram = (float*)d_ws;               // B_*10 floats
    float* prob = gram + 256;                 // B_*4 floats (aligned slot)

    zero_kernel<<<1, 256, 0, stream>>>(gram, B_ * 10);
    gram_kernel<<<dim3(B_ * 64), 256, 0, stream>>>(x, gram);
    attn_kernel<<<1, 32, 0, stream>>>(gram, prob);
    gemm_kernel<<<dim3(B_ * (HW / NB)), 256, 0, stream>>>(x, w, prob, out);
    inorm_kernel<<<dim3(B_ * COUT), 256, 0, stream>>>(out);
}